// TtMambaSSM_41901700940188
// MI455X (gfx1250) — compile-verified
//
#include <hip/hip_runtime.h>
#include <math.h>

typedef __attribute__((ext_vector_type(2))) float v2f;
typedef __attribute__((ext_vector_type(4))) float v4f;
typedef __attribute__((ext_vector_type(8))) float v8f;

#define U_DIM 256
#define H_DIM 5120
#define R_DIM 640
#define N_ST  16
#define STATE (H_DIM * N_ST)          // 81920 per u

// ---------------------------------------------------------------------------
// Fragment helpers for V_WMMA_F32_16X16X4_F32 (wave32).
// A (16x4, MxK): lane L -> m = L%16, k = 2*(L/16) + {0,1} in {x,y}
// B (4x16, KxN): lane L -> n = L%16, k = 2*(L/16) + {0,1} in {x,y}
// C/D (16x16):   lane L -> n = L%16, VGPR v -> m = v + 8*(L/16)
// ---------------------------------------------------------------------------

// tmp[256,640] = x[256,5120] @ W_dt_rank[5120,640]
__global__ __launch_bounds__(32) void k_gemm_tmp(const float* __restrict__ x,
                                                 const float* __restrict__ Wdr,
                                                 float* __restrict__ tmp) {
    const int lane = threadIdx.x;
    const int half = lane >> 4;
    const int lm   = lane & 15;
    const int tn   = blockIdx.x;          // 0..39
    const int tm   = blockIdx.y;          // 0..15
    const int row  = tm * 16 + lm;
    const int col  = tn * 16 + lm;
    const float* arow = x + (size_t)row * H_DIM;

    v8f acc0 = {}; v8f acc1 = {};
    for (int k = 0; k < H_DIM; k += 8) {
        int kb0 = k + half * 2;
        v2f a0 = *(const v2f*)(arow + kb0);
        v2f b0; b0.x = Wdr[(size_t)kb0 * R_DIM + col];
                b0.y = Wdr[(size_t)(kb0 + 1) * R_DIM + col];
        acc0 = __builtin_amdgcn_wmma_f32_16x16x4_f32(false, a0, false, b0,
                                                     (short)0, acc0, false, false);
        int kb1 = k + 4 + half * 2;
        v2f a1 = *(const v2f*)(arow + kb1);
        v2f b1; b1.x = Wdr[(size_t)kb1 * R_DIM + col];
                b1.y = Wdr[(size_t)(kb1 + 1) * R_DIM + col];
        acc1 = __builtin_amdgcn_wmma_f32_16x16x4_f32(false, a1, false, b1,
                                                     (short)0, acc1, false, false);
    }
    v8f acc = acc0 + acc1;
#pragma unroll
    for (int v = 0; v < 8; ++v) {
        int m = v + half * 8;
        tmp[(size_t)(tm * 16 + m) * R_DIM + col] = acc[v];
    }
}

// Bv[256,16] = x @ W_B ; Cv[256,16] = x @ W_C   (single N-tile, two accumulators)
__global__ __launch_bounds__(32) void k_gemm_bc(const float* __restrict__ x,
                                                const float* __restrict__ WB,
                                                const float* __restrict__ WC,
                                                float* __restrict__ Bv,
                                                float* __restrict__ Cv) {
    const int lane = threadIdx.x;
    const int half = lane >> 4;
    const int lm   = lane & 15;
    const int tm   = blockIdx.x;          // 0..15
    const int row  = tm * 16 + lm;
    const float* arow = x + (size_t)row * H_DIM;

    v8f accB = {}; v8f accC = {};
    for (int k = 0; k < H_DIM; k += 4) {
        int kb = k + half * 2;
        v2f a = *(const v2f*)(arow + kb);
        v2f bB; bB.x = WB[kb * N_ST + lm]; bB.y = WB[(kb + 1) * N_ST + lm];
        v2f bC; bC.x = WC[kb * N_ST + lm]; bC.y = WC[(kb + 1) * N_ST + lm];
        accB = __builtin_amdgcn_wmma_f32_16x16x4_f32(false, a, false, bB,
                                                     (short)0, accB, false, false);
        accC = __builtin_amdgcn_wmma_f32_16x16x4_f32(false, a, false, bC,
                                                     (short)0, accC, false, false);
    }
#pragma unroll
    for (int v = 0; v < 8; ++v) {
        int m = v + half * 8;
        Bv[(tm * 16 + m) * N_ST + lm] = accB[v];
        Cv[(tm * 16 + m) * N_ST + lm] = accC[v];
    }
}

__device__ __forceinline__ float softplus_f(float v) {
    // numerically stable log(1+exp(v))
    return (v > 0.0f) ? (v + log1pf(expf(-v))) : log1pf(expf(v));
}

// dt[256,5120] = softplus(tmp[256,640] @ W_dt[640,5120] + b_dt)
__global__ __launch_bounds__(32) void k_gemm_dt(const float* __restrict__ tmp,
                                                const float* __restrict__ Wdt,
                                                const float* __restrict__ bdt,
                                                float* __restrict__ dt) {
    const int lane = threadIdx.x;
    const int half = lane >> 4;
    const int lm   = lane & 15;
    const int tn   = blockIdx.x;          // 0..319
    const int tm   = blockIdx.y;          // 0..15
    const int row  = tm * 16 + lm;
    const int col  = tn * 16 + lm;
    const float* arow = tmp + (size_t)row * R_DIM;

    v8f acc0 = {}; v8f acc1 = {};
    for (int k = 0; k < R_DIM; k += 8) {
        int kb0 = k + half * 2;
        v2f a0 = *(const v2f*)(arow + kb0);
        v2f b0; b0.x = Wdt[(size_t)kb0 * H_DIM + col];
                b0.y = Wdt[(size_t)(kb0 + 1) * H_DIM + col];
        acc0 = __builtin_amdgcn_wmma_f32_16x16x4_f32(false, a0, false, b0,
                                                     (short)0, acc0, false, false);
        int kb1 = k + 4 + half * 2;
        v2f a1 = *(const v2f*)(arow + kb1);
        v2f b1; b1.x = Wdt[(size_t)kb1 * H_DIM + col];
                b1.y = Wdt[(size_t)(kb1 + 1) * H_DIM + col];
        acc1 = __builtin_amdgcn_wmma_f32_16x16x4_f32(false, a1, false, b1,
                                                     (short)0, acc1, false, false);
    }
    v8f acc = acc0 + acc1;
    const float bias = bdt[col];
#pragma unroll
    for (int v = 0; v < 8; ++v) {
        int m = v + half * 8;
        dt[(size_t)(tm * 16 + m) * H_DIM + col] = softplus_f(acc[v] + bias);
    }
}

// Fused streaming pass: one thread per (u, hh) group of 16 states.
// h_new[n] = exp(A)*h + dt[u,hh]*x[u,hh]*B[u, hh/320]
// hs3: each h_new duplicated x2 (32 floats); out[u,hh] = 2 * dot(h_new, C[u,:])
__global__ __launch_bounds__(256) void k_ssm(const float* __restrict__ A,
                                             const float* __restrict__ h,
                                             const float* __restrict__ x,
                                             const float* __restrict__ dt,
                                             const float* __restrict__ Bv,
                                             const float* __restrict__ Cv,
                                             float* __restrict__ hs3,
                                             float* __restrict__ outv) {
    const int t = blockIdx.x * blockDim.x + threadIdx.x;   // 0 .. U*H-1
    if (t >= U_DIM * H_DIM) return;
    const int u  = t / H_DIM;
    const int hh = t - u * H_DIM;

    const float c = dt[t] * x[t] * Bv[u * N_ST + (hh / (H_DIM / N_ST))];

    const size_t sbase = (size_t)u * STATE + (size_t)hh * N_ST;
    const v4f* Ap = (const v4f*)(A + sbase);
    const v4f* hp = (const v4f*)(h + sbase);
    v4f* op = (v4f*)(hs3 + (size_t)u * (size_t)(H_DIM * 2 * N_ST)
                         + (size_t)hh * (2 * N_ST));
    const float* Cu = Cv + u * N_ST;

    float dot = 0.0f;
#pragma unroll
    for (int j = 0; j < 4; ++j) {
        v4f a4 = __builtin_nontemporal_load(Ap + j);
        v4f h4 = __builtin_nontemporal_load(hp + j);
        v4f hn;
        hn.x = expf(a4.x) * h4.x + c;
        hn.y = expf(a4.y) * h4.y + c;
        hn.z = expf(a4.z) * h4.z + c;
        hn.w = expf(a4.w) * h4.w + c;
        dot += hn.x * Cu[4 * j + 0] + hn.y * Cu[4 * j + 1]
             + hn.z * Cu[4 * j + 2] + hn.w * Cu[4 * j + 3];
        v4f o0 = { hn.x, hn.x, hn.y, hn.y };
        v4f o1 = { hn.z, hn.z, hn.w, hn.w };
        __builtin_nontemporal_store(o0, op + 2 * j);
        __builtin_nontemporal_store(o1, op + 2 * j + 1);
    }
    outv[t] = 2.0f * dot;
}

extern "C" void kernel_launch(void* const* d_in, const int* in_sizes, int n_in,
                              void* d_out, int out_size, void* d_ws, size_t ws_size,
                              hipStream_t stream) {
    const float* x   = (const float*)d_in[0];
    const float* A   = (const float*)d_in[1];
    const float* h   = (const float*)d_in[2];
    const float* Wdr = (const float*)d_in[3];
    const float* Wdt = (const float*)d_in[4];
    const float* bdt = (const float*)d_in[5];
    const float* WB  = (const float*)d_in[6];
    const float* WC  = (const float*)d_in[7];

    // workspace layout (bytes, 256B-aligned blocks)
    char* ws = (char*)d_ws;
    float* tmp = (float*)(ws);                                   // 256*640
    float* dt  = (float*)(ws + ((size_t)U_DIM * R_DIM * 4));     // 256*5120
    float* Bv  = (float*)(ws + ((size_t)U_DIM * R_DIM * 4)
                             + ((size_t)U_DIM * H_DIM * 4));     // 256*16
    float* Cv  = Bv + (size_t)U_DIM * N_ST;                      // 256*16

    float* hs3  = (float*)d_out;                                  // 256*5120*32
    float* outv = hs3 + (size_t)U_DIM * H_DIM * 2 * N_ST;         // 256*5120

    k_gemm_tmp<<<dim3(R_DIM / 16, U_DIM / 16), 32, 0, stream>>>(x, Wdr, tmp);
    k_gemm_bc <<<U_DIM / 16, 32, 0, stream>>>(x, WB, WC, Bv, Cv);
    k_gemm_dt <<<dim3(H_DIM / 16, U_DIM / 16), 32, 0, stream>>>(tmp, Wdt, bdt, dt);

    const int total = U_DIM * H_DIM;
    k_ssm<<<(total + 255) / 256, 256, 0, stream>>>(A, h, x, dt, Bv, Cv, hs3, outv);
}